// JpegLayer_71287867179496
// MI455X (gfx1250) — compile-verified
//
#include <hip/hip_runtime.h>
#include <math.h>

typedef __attribute__((ext_vector_type(2))) float v2f;
typedef __attribute__((ext_vector_type(8))) float v8f;
typedef __attribute__((ext_vector_type(4))) unsigned int u32x4;
typedef __attribute__((ext_vector_type(8))) int i32x8;
typedef __attribute__((ext_vector_type(4))) int i32x4;

#define JP_HALF 0.5019607843137255f   // 128/255
#define LROW 17                       // padded LDS row pitch (avoids 16-row bank aliasing)

#if __has_builtin(__builtin_amdgcn_tensor_load_to_lds) && __has_builtin(__builtin_amdgcn_s_wait_tensorcnt)
#define USE_TDM 1
#else
#define USE_TDM 0
#endif

// ---------------- DCT basis (compile-time table, replaces runtime cosf) ----------
// D[i][j] of the 8x8 DCT-II matrix as built by _dctmtx()
__device__ __constant__ float DCT8[8][8] = {
    { 0.35355339059f,  0.35355339059f,  0.35355339059f,  0.35355339059f,
      0.35355339059f,  0.35355339059f,  0.35355339059f,  0.35355339059f },
    { 0.49039264020f,  0.41573480615f,  0.27778511651f,  0.09754516101f,
     -0.09754516101f, -0.27778511651f, -0.41573480615f, -0.49039264020f },
    { 0.46193976626f,  0.19134171618f, -0.19134171618f, -0.46193976626f,
     -0.46193976626f, -0.19134171618f,  0.19134171618f,  0.46193976626f },
    { 0.41573480615f, -0.09754516101f, -0.49039264020f, -0.27778511651f,
      0.27778511651f,  0.49039264020f,  0.09754516101f, -0.41573480615f },
    { 0.35355339059f, -0.35355339059f, -0.35355339059f,  0.35355339059f,
      0.35355339059f, -0.35355339059f, -0.35355339059f,  0.35355339059f },
    { 0.27778511651f, -0.49039264020f,  0.09754516101f,  0.41573480615f,
     -0.41573480615f, -0.09754516101f,  0.49039264020f, -0.27778511651f },
    { 0.19134171618f, -0.46193976626f,  0.46193976626f, -0.19134171618f,
     -0.19134171618f,  0.46193976626f, -0.46193976626f,  0.19134171618f },
    { 0.09754516101f, -0.27778511651f,  0.41573480615f, -0.49039264020f,
      0.49039264020f, -0.41573480615f,  0.27778511651f, -0.09754516101f }
};

// 16x16 block-diagonal DD = diag(D, D): one 16x16 tile = 2x2 independent 8x8 blocks
__device__ __forceinline__ float DDe(int r, int c) {
    return ((r >> 3) == (c >> 3)) ? DCT8[r & 7][c & 7] : 0.0f;
}

// ---------------- kernel 0: zero accumulators ----------------
__global__ void k_zero_acc(float* acc) {
    if (threadIdx.x < 8) acc[threadIdx.x] = 0.0f;
}

// ---------------- kernel 1: RGB -> YCbCr + 2x2 chroma subsample ----------------
__global__ __launch_bounds__(256) void k_rgb2ycc(const float* __restrict__ x,
                                                 float* __restrict__ yPl,
                                                 float* __restrict__ cbPl,
                                                 float* __restrict__ crPl,
                                                 int nimg)
{
    int idx = blockIdx.x * blockDim.x + threadIdx.x;
    int total = nimg * 256 * 256;
    if (idx >= total) return;
    int img = idx / (256 * 256);
    int rem = idx - img * (256 * 256);
    int wc = rem >> 8;
    int hc = rem & 255;

    const float* R = x + (size_t)img * 3 * 512 * 512;
    const float* G = R + 512 * 512;
    const float* B = G + 512 * 512;
    float* yo = yPl + (size_t)img * 512 * 512;

    float cb = 0.f, cr = 0.f;
    #pragma unroll
    for (int dw = 0; dw < 2; ++dw)
        #pragma unroll
        for (int dh = 0; dh < 2; ++dh) {
            int off = (2 * wc + dw) * 512 + (2 * hc + dh);
            float r = R[off], g = G[off], b = B[off];
            yo[off] = 0.299f * r + 0.587f * g + 0.114f * b - JP_HALF;   // Y: (+0 then -HALF)
            cb += -0.168735892f * r - 0.331264108f * g + 0.5f * b;      // Cb/Cr: +HALF-HALF = 0
            cr += 0.5f * r - 0.418687589f * g - 0.081312411f * b;
        }
    cbPl[idx] = 0.25f * cb;
    crPl[idx] = 0.25f * cr;
}

// ---------------- kernel 2: per-plane DCT -> quant -> dequant -> IDCT (in place) ----
// One wave owns one 16x16 tile (= 2x2 JPEG 8x8 blocks). Forward: F = DD*T*DD^T,
// inverse: R = DD^T*dec*DD, each as 4 chained V_WMMA_F32_16X16X4_F32.
// Tile load goes through the Tensor Data Mover (global -> LDS, HW row padding to
// pitch 17 DWORDs), tracked on TENSORcnt.
__global__ __launch_bounds__(256) void k_dct_quant_idct(
    float* __restrict__ plane, const float* __restrict__ qt,
    float* __restrict__ absAcc, float* __restrict__ zeroAcc,
    int nTiles, int tilesW, int tilesH, int W, int H)
{
    __shared__ float ldsA[8][16 * LROW];
    __shared__ float ldsB[8][16 * LROW];

    const int lane = threadIdx.x & 31;
    const int wave = threadIdx.x >> 5;
    const int hi   = lane >> 4;         // which lane-half: selects K pair / M half
    const int M    = lane & 15;

    float* A  = ldsA[wave];
    float* Bm = ldsB[wave];

    // Per-lane constant fragments of DD.
    // fD[kk]  : A-fragment of DD   (== B-fragment of DD^T)
    // fDt[kk] : A-fragment of DD^T (== B-fragment of DD)
    v2f fD[4], fDt[4];
    #pragma unroll
    for (int kk = 0; kk < 4; ++kk) {
        int k = kk * 4 + 2 * hi;
        fD[kk].x  = DDe(M, k);     fD[kk].y  = DDe(M, k + 1);
        fDt[kk].x = DDe(k, M);     fDt[kk].y = DDe(k + 1, M);
    }
    // D-fragment element (v, lane) maps to block position (row v, col lane&7).
    // Hoist the expensive fp32 divide out of the tile loop: quant = rintf(f * (1/q)).
    float qv[8], rq[8];
    #pragma unroll
    for (int v = 0; v < 8; ++v) {
        qv[v] = qt[v * 8 + (lane & 7)];
        rq[v] = 1.0f / qv[v];
    }

    float absSum = 0.0f, zeroSum = 0.0f;
    const int wavesPerBlock = blockDim.x >> 5;
    const int waveStride = gridDim.x * wavesPerBlock;

#if USE_TDM
    const unsigned int ldsOff = (unsigned int)(uintptr_t)(void*)A;  // low 32 bits = LDS offset
#endif

    for (int base = blockIdx.x * wavesPerBlock; base < nTiles; base += waveStride) {
        int  t     = base + wave;
        bool valid = (t < nTiles);
        int  tt    = valid ? t : (nTiles - 1);
        int  img   = tt / (tilesW * tilesH);
        int  rem   = tt - img * (tilesW * tilesH);
        int  ti    = rem / tilesH;
        int  tj    = rem - ti * tilesH;
        float* g = plane + (size_t)img * W * H + (size_t)(ti * 16) * H + tj * 16;

        // prefetch next tile (global_prefetch_b8)
        int tn = t + waveStride;
        if (tn < nTiles) {
            int img2 = tn / (tilesW * tilesH);
            int rem2 = tn - img2 * (tilesW * tilesH);
            int ti2  = rem2 / tilesH;
            int tj2  = rem2 - ti2 * tilesH;
            __builtin_prefetch(plane + (size_t)img2 * W * H + (size_t)(ti2 * 16) * H + tj2 * 16, 0, 1);
        }

        // phase 0: tile -> LDS
#if USE_TDM
        {
            unsigned long long ga = (unsigned long long)(uintptr_t)g;
            // D# group 0: count=1 | lds_addr | global_addr[56:0] | type=2
            u32x4 g0;
            g0.x = 1u;                                   // count=1, user mode, no gather
            g0.y = ldsOff;                               // LDS byte address of tile
            g0.z = (unsigned int)ga;                     // global_addr[31:0]
            g0.w = ((unsigned int)(ga >> 32) & 0x01FFFFFFu) | 0x80000000u; // addr[56:32] | type=2
            // D# group 1: data_size=4B, pad_enable, pad every 16 DWORDs by 1 DWORD (pitch 17),
            //             huge tensor dims (no OOB clip), tile 16x16, dim0 stride = H elements.
            const int TD = 1 << 20;                      // tensor_dim0 = tensor_dim1 = 2^20
            i32x8 g1;
            g1[0] = (int)((2u << 16) | (1u << 20) | (3u << 22) | (0u << 25)); // ds=4B,pad on,intv=16DW,amt=1DW
            g1[1] = (int)((TD & 0xFFFF) << 16);          // abar=0 | tensor_dim0[15:0]
            g1[2] = (int)((TD >> 16) | ((TD & 0xFFFF) << 16));   // tensor_dim0[31:16] | tensor_dim1[15:0]
            g1[3] = (int)((TD >> 16) | (16u << 16));     // tensor_dim1[31:16] | tile_dim0=16
            g1[4] = 16;                                  // tile_dim1=16 | tile_dim2=0
            g1[5] = H;                                   // tensor_dim0_stride[31:0] = H
            g1[6] = 0;                                   // stride0[47:32] | stride1[15:0]
            g1[7] = 0;
            i32x4 gz4 = {0, 0, 0, 0};
            i32x8 gz8 = {0, 0, 0, 0, 0, 0, 0, 0};
            // 6-arg form (clang-23 / therock-10.0): (g0, g1, g2, g3, g4, cpol)
            __builtin_amdgcn_tensor_load_to_lds(g0, g1, gz4, gz4, gz8, 0);
            __builtin_amdgcn_s_wait_tensorcnt(0);
        }
#else
        #pragma unroll
        for (int v = 0; v < 8; ++v) {
            int r = v + 8 * hi;
            A[r * LROW + M] = g[r * H + M];
        }
#endif
        __syncthreads();

        // phase 1: S = DD * T
        v8f s = {};
        #pragma unroll
        for (int kk = 0; kk < 4; ++kk) {
            int kb = kk * 4 + 2 * hi;
            v2f b; b.x = A[kb * LROW + M]; b.y = A[(kb + 1) * LROW + M];
            s = __builtin_amdgcn_wmma_f32_16x16x4_f32(false, fD[kk], false, b, (short)0, s, false, false);
        }
        #pragma unroll
        for (int v = 0; v < 8; ++v) Bm[(v + 8 * hi) * LROW + M] = s[v];
        __syncthreads();

        // phase 2: F = S * DD^T
        v8f f = {};
        #pragma unroll
        for (int kk = 0; kk < 4; ++kk) {
            int kb = kk * 4 + 2 * hi;
            v2f a; a.x = Bm[M * LROW + kb]; a.y = Bm[M * LROW + kb + 1];
            f = __builtin_amdgcn_wmma_f32_16x16x4_f32(false, a, false, fD[kk], (short)0, f, false, false);
        }

        // phase 3: quantize / dequantize + stats; write dec to LDS in D-frag layout
        #pragma unroll
        for (int v = 0; v < 8; ++v) {
            float quant = rintf(f[v] * rq[v]);   // RNE rounding, reciprocal hoisted
            if (valid) {
                absSum  += fabsf(quant);
                zeroSum += (quant == 0.0f) ? 1.0f : 0.0f;
            }
            A[(v + 8 * hi) * LROW + M] = quant * qv[v];
        }
        __syncthreads();

        // phase 4: R1 = DD^T * dec
        v8f r1 = {};
        #pragma unroll
        for (int kk = 0; kk < 4; ++kk) {
            int kb = kk * 4 + 2 * hi;
            v2f b; b.x = A[kb * LROW + M]; b.y = A[(kb + 1) * LROW + M];
            r1 = __builtin_amdgcn_wmma_f32_16x16x4_f32(false, fDt[kk], false, b, (short)0, r1, false, false);
        }
        #pragma unroll
        for (int v = 0; v < 8; ++v) Bm[(v + 8 * hi) * LROW + M] = r1[v];
        __syncthreads();

        // phase 5: R = R1 * DD
        v8f r = {};
        #pragma unroll
        for (int kk = 0; kk < 4; ++kk) {
            int kb = kk * 4 + 2 * hi;
            v2f a; a.x = Bm[M * LROW + kb]; a.y = Bm[M * LROW + kb + 1];
            r = __builtin_amdgcn_wmma_f32_16x16x4_f32(false, a, false, fDt[kk], (short)0, r, false, false);
        }

        // phase 6: store reconstructed tile (in place)
        if (valid) {
            #pragma unroll
            for (int v = 0; v < 8; ++v) {
                int rr = v + 8 * hi;
                g[rr * H + M] = r[v];
            }
        }
        __syncthreads();
    }

    // wave reduction of stats, one atomic per wave
    #pragma unroll
    for (int off = 16; off > 0; off >>= 1) {
        absSum  += __shfl_down(absSum,  off, 32);
        zeroSum += __shfl_down(zeroSum, off, 32);
    }
    if (lane == 0) {
        atomicAdd(absAcc,  absSum);
        atomicAdd(zeroAcc, zeroSum);
    }
}

// ---------------- kernel 3: upsample chroma + YCbCr -> RGB + scalars --------------
__global__ __launch_bounds__(256) void k_ycc2rgb(const float* __restrict__ yPl,
                                                 const float* __restrict__ cbPl,
                                                 const float* __restrict__ crPl,
                                                 float* __restrict__ out,
                                                 const float* __restrict__ acc,
                                                 int nimg, int npix)
{
    int idx = blockIdx.x * blockDim.x + threadIdx.x;
    int total = nimg * 256 * 256;
    if (idx < total) {
        int img = idx / (256 * 256);
        int rem = idx - img * (256 * 256);
        int wc = rem >> 8;
        int hc = rem & 255;
        const float* yi = yPl + (size_t)img * 512 * 512;
        float cb = cbPl[idx], cr = crPl[idx];
        float* o = out + (size_t)img * 3 * 512 * 512;
        #pragma unroll
        for (int dw = 0; dw < 2; ++dw)
            #pragma unroll
            for (int dh = 0; dh < 2; ++dh) {
                int off = (2 * wc + dw) * 512 + (2 * hc + dh);
                float Y = yi[off] + JP_HALF;   // Y gets +HALF, chroma nets to 0
                o[off]                 = Y + 1.402f * cr;
                o[512 * 512 + off]     = Y - 0.344136286f * cb - 0.714136286f * cr;
                o[2 * 512 * 512 + off] = Y + 1.772f * cb;
            }
    }
    if (blockIdx.x == 0 && threadIdx.x == 0) {
        float Ny = (float)nimg * 512.f * 512.f;
        float Nc = (float)nimg * 256.f * 256.f;
        out[npix]     = acc[0] / Ny + acc[1] / Nc + acc[2] / Nc;  // sum of per-channel means
        out[npix + 1] = acc[5] / Nc;                              // nzeros: Cr only (ref overwrites)
    }
}

// ---------------- host launcher ----------------
extern "C" void kernel_launch(void* const* d_in, const int* in_sizes, int n_in,
                              void* d_out, int out_size, void* d_ws, size_t ws_size,
                              hipStream_t stream)
{
    const float* x  = (const float*)d_in[0];
    const float* qt = (const float*)d_in[1];
    float* out = (float*)d_out;
    int nimg = in_sizes[0] / (3 * 512 * 512);

    // workspace layout: [8 float accumulators | Y plane | Cb plane | Cr plane]
    float* acc  = (float*)d_ws;
    float* yPl  = (float*)((char*)d_ws + 256);
    float* cbPl = yPl  + (size_t)nimg * 512 * 512;
    float* crPl = cbPl + (size_t)nimg * 256 * 256;

    int npix = nimg * 3 * 512 * 512;

    k_zero_acc<<<1, 32, 0, stream>>>(acc);

    int total   = nimg * 256 * 256;
    int cblocks = (total + 255) / 256;
    k_rgb2ycc<<<cblocks, 256, 0, stream>>>(x, yPl, cbPl, crPl, nimg);

    int ntY = nimg * 32 * 32;   // (512/16)^2 tiles per image
    int ntC = nimg * 16 * 16;   // (256/16)^2 tiles per image
    k_dct_quant_idct<<<2048, 256, 0, stream>>>(yPl,  qt, acc + 0, acc + 3, ntY, 32, 32, 512, 512);
    k_dct_quant_idct<<<1024, 256, 0, stream>>>(cbPl, qt, acc + 1, acc + 4, ntC, 16, 16, 256, 256);
    k_dct_quant_idct<<<1024, 256, 0, stream>>>(crPl, qt, acc + 2, acc + 5, ntC, 16, 16, 256, 256);

    k_ycc2rgb<<<cblocks, 256, 0, stream>>>(yPl, cbPl, crPl, out, acc, nimg, npix);
}